// SAGE_19567871000655
// MI455X (gfx1250) — compile-verified
//
#include <hip/hip_runtime.h>

typedef __attribute__((ext_vector_type(2))) float v2f;
typedef __attribute__((ext_vector_type(8))) float v8f;

#define D_IN 128

// ---------------------------------------------------------------------------
// Edge scatter: one 32-lane group per edge. Each lane owns a float4 chunk of
// the 128-float feature row: gather feat[src], global_atomic_add_f32 into
// agg[dst]. Lane 0 optionally counts degree. Random traffic stays in the
// 192MB L2 (whole working set ~190MB), so atomics resolve on-chip.
// ---------------------------------------------------------------------------
__global__ void __launch_bounds__(256)
sage_scatter(const float* __restrict__ feat, const int* __restrict__ src,
             const int* __restrict__ dst, float* __restrict__ agg,
             float* __restrict__ deg, int E) {
  int gid = blockIdx.x * blockDim.x + threadIdx.x;
  int e = gid >> 5;
  if (e >= E) return;
  int lane = gid & 31;
  int s = src[e];
  int d = dst[e];
  const float4 v = *(const float4*)(feat + (size_t)s * D_IN + lane * 4);
  float* b = agg + (size_t)d * D_IN + lane * 4;
  atomicAdd(b + 0, v.x);
  atomicAdd(b + 1, v.y);
  atomicAdd(b + 2, v.z);
  atomicAdd(b + 3, v.w);
  if (deg != nullptr && lane == 0) atomicAdd(deg + d, 1.0f);
}

// ---------------------------------------------------------------------------
// Fused SAGE linear: out[64 rows x OUTC] = (agg/deg) @ Wl + x @ Wr + bias,
// optional ReLU. fp32 WMMA 16x16x4, K=128.
//
// Block = 4 row-tiles of 16 rows; one wave per 16-column slice. B (weight)
// fragments are loaded ONCE per k-step and reused across the 4 row-tiles
// (4x less L2 weight traffic, 2 WMMA per weight load).
//
// VGPR layouts per CDNA5 ISA 7.12.2 (wave32):
//   A (16x4 f32):  lane = 16*h + m (h=0/1). v0 = A[m][2h],   v1 = A[m][2h+1]
//   B (4x16 f32):  v0 = B[2h][n],  v1 = B[2h+1][n],  n = lane&15
//   C/D (16x16):   c[v] = C[v + 8h][n]
// ---------------------------------------------------------------------------
template <int OUTC, bool RELU>
__global__ void __launch_bounds__(OUTC * 2)
sage_gemm(const float* __restrict__ agg, const float* __restrict__ deg,
          const float* __restrict__ x, const float* __restrict__ Wl,
          const float* __restrict__ Wr, const float* __restrict__ bias,
          float* __restrict__ out, int N) {
  constexpr int RT = 4;             // row tiles per block
  constexpr int ROWS = 16 * RT;     // 64 rows per block
  constexpr int LDS_LD = D_IN + 4;  // 132 % 64 banks = 4 -> conflict-free rows
  __shared__ float sAgg[ROWS * LDS_LD];  // 33.8 KB
  __shared__ float sX[ROWS * LDS_LD];    // 33.8 KB
  __shared__ float sRdeg[ROWS];

  const int row0 = blockIdx.x * ROWS;

  // Stage the 64x128 A-tiles (agg + x) into LDS with float4 moves.
  for (int i = threadIdx.x; i < ROWS * (D_IN / 4); i += blockDim.x) {
    int r = i >> 5;            // 32 float4 per row
    int cq = (i & 31) * 4;
    int rr = row0 + r;
    if (rr >= N) rr = N - 1;   // clamp; stores are guarded below
    *(float4*)(sAgg + r * LDS_LD + cq) =
        *(const float4*)(agg + (size_t)rr * D_IN + cq);
    *(float4*)(sX + r * LDS_LD + cq) =
        *(const float4*)(x + (size_t)rr * D_IN + cq);
  }
  for (int i = threadIdx.x; i < ROWS; i += blockDim.x) {
    int rr = row0 + i;
    if (rr >= N) rr = N - 1;
    sRdeg[i] = 1.0f / fmaxf(deg[rr], 1.0f);
  }
  __syncthreads();

  const int lane = threadIdx.x & 31;
  const int wave = threadIdx.x >> 5;
  const int n0 = wave * 16;          // this wave's output-column slice
  const int m = lane & 15;           // A row / B,C column within tile
  const int kq = (lane >> 4) * 2;    // K sub-offset from lane half

  // Per-row-tile accumulators, initialized with bias (each C row = bias[col]).
  v8f acc[RT];
  const float bv = bias[n0 + m];
#pragma unroll
  for (int t = 0; t < RT; ++t) {
#pragma unroll
    for (int v = 0; v < 8; ++v) acc[t][v] = bv;
  }

  float rd[RT];
  const float* sa[RT];
  const float* sx[RT];
#pragma unroll
  for (int t = 0; t < RT; ++t) {
    rd[t] = sRdeg[t * 16 + m];
    sa[t] = sAgg + (t * 16 + m) * LDS_LD;
    sx[t] = sX + (t * 16 + m) * LDS_LD;
  }

#pragma unroll
  for (int k0 = 0; k0 < D_IN; k0 += 4) {
    const int k = k0 + kq;
    // Weight fragments: loaded once, reused by all 4 row-tiles.
    v2f bl, br;
    bl[0] = Wl[k * OUTC + n0 + m];
    bl[1] = Wl[(k + 1) * OUTC + n0 + m];
    br[0] = Wr[k * OUTC + n0 + m];
    br[1] = Wr[(k + 1) * OUTC + n0 + m];
#pragma unroll
    for (int t = 0; t < RT; ++t) {
      v2f a;
      // mean-aggregated path: (agg * 1/deg) @ Wl
      a[0] = sa[t][k] * rd[t];
      a[1] = sa[t][k + 1] * rd[t];
      acc[t] = __builtin_amdgcn_wmma_f32_16x16x4_f32(false, a, false, bl,
                                                     (short)0, acc[t], false,
                                                     false);
      // self path: x @ Wr
      a[0] = sx[t][k];
      a[1] = sx[t][k + 1];
      acc[t] = __builtin_amdgcn_wmma_f32_16x16x4_f32(false, a, false, br,
                                                     (short)0, acc[t], false,
                                                     false);
    }
  }

#pragma unroll
  for (int t = 0; t < RT; ++t) {
#pragma unroll
    for (int v = 0; v < 8; ++v) {
      int row = row0 + t * 16 + v + ((lane >> 4) << 3);
      if (row < N) {
        float val = acc[t][v];
        if (RELU) val = fmaxf(val, 0.0f);
        out[(size_t)row * OUTC + n0 + m] = val;
      }
    }
  }
}

// ---------------------------------------------------------------------------
// Launch: memset scratch -> scatter1 -> gemm1(relu) -> scatter2 -> gemm2
// Workspace layout (floats): agg1[N*128] | h[N*128] | agg2[N*128] | deg[N]
// ---------------------------------------------------------------------------
extern "C" void kernel_launch(void* const* d_in, const int* in_sizes, int n_in,
                              void* d_out, int out_size, void* d_ws,
                              size_t ws_size, hipStream_t stream) {
  const float* x = (const float*)d_in[0];
  const int* ei = (const int*)d_in[1];
  const float* W1l = (const float*)d_in[2];
  const float* b1 = (const float*)d_in[3];
  const float* W1r = (const float*)d_in[4];
  const float* W2l = (const float*)d_in[5];
  const float* b2 = (const float*)d_in[6];
  const float* W2r = (const float*)d_in[7];
  float* out = (float*)d_out;

  const int N = in_sizes[0] / D_IN;
  const int E = in_sizes[1] / 2;
  const int* src = ei;      // edge_index[0, :]
  const int* dst = ei + E;  // edge_index[1, :]

  char* ws = (char*)d_ws;
  const size_t featBytes = (size_t)N * D_IN * sizeof(float);
  float* agg1 = (float*)(ws);
  float* h = (float*)(ws + featBytes);
  float* agg2 = (float*)(ws + 2 * featBytes);
  float* deg = (float*)(ws + 3 * featBytes);

  hipMemsetAsync(agg1, 0, featBytes, stream);
  hipMemsetAsync(agg2, 0, featBytes, stream);
  hipMemsetAsync(deg, 0, (size_t)N * sizeof(float), stream);

  const int scatterThreads = E * 32;
  const int scatterBlocks = (scatterThreads + 255) / 256;
  const int tiles = (N + 63) / 64;  // 64 rows per block

  // Layer 1
  sage_scatter<<<scatterBlocks, 256, 0, stream>>>(x, src, dst, agg1, deg, E);
  sage_gemm<128, true>
      <<<tiles, 256, 0, stream>>>(agg1, deg, x, W1l, W1r, b1, h, N);

  // Layer 2 (dropout is identity in eval mode)
  sage_scatter<<<scatterBlocks, 256, 0, stream>>>(h, src, dst, agg2, nullptr,
                                                  E);
  sage_gemm<64, false>
      <<<tiles, 128, 0, stream>>>(agg2, deg, h, W2l, W2r, b2, out, N);
}